// SpectralTCNVQVAE_24781961298457
// MI455X (gfx1250) — compile-verified
//
#include <hip/hip_runtime.h>

typedef __attribute__((ext_vector_type(2))) float v2f;
typedef __attribute__((ext_vector_type(8))) float v8f;

#define HIDDEN   256
#define NBANDS   200
#define MTILES   13          // ceil(200/16): tiles cover rows 0..207
#define PADROWS  210         // padded LDS rows 0..209 (t = p-1 for p in 1..200)
#define KBLOCKS  192         // K = 3*256 = 768, 4 per V_WMMA_F32_16X16X4_F32

// ------------------------------------------------------------------
// Kernel 1: x_mean over the trailing 64x64 = 4096 elements (HBM-bound)
// ------------------------------------------------------------------
__global__ void k_xmean(const float* __restrict__ x, float* __restrict__ xmean) {
  __shared__ float red[256];
  const int row = blockIdx.x;                     // b*200 + band
  const float4* p = (const float4*)(x + (size_t)row * 4096);
  float s = 0.f;
  #pragma unroll
  for (int i = 0; i < 4; ++i) {
    float4 v = p[threadIdx.x + i * 256];
    s += v.x + v.y + v.z + v.w;
  }
  red[threadIdx.x] = s;
  __syncthreads();
  for (int off = 128; off > 0; off >>= 1) {
    if (threadIdx.x < off) red[threadIdx.x] += red[threadIdx.x + off];
    __syncthreads();
  }
  if (threadIdx.x == 0) xmean[row] = red[0] * (1.0f / 4096.0f);
}

// ------------------------------------------------------------------
// f32 WMMA GEMM core: one wave computes a 16-wide oc tile over all 13
// M-tiles, K = 768.  A comes from LDS (implicit im2col), B from global
// (L2-resident weights), reused across the 13 M-tiles.
// Fragment layout (ISA 7.12.2, f32):
//   A 16x4 : lane L: M = L&15, v[j] = K (j + 2*(L>>4))
//   B 4x16 : lane L: N = L&15, v[j] = K (j + 2*(L>>4))
//   C 16x16: lane L: v[r] = row (r + 8*(L>>4)), col (L&15)
// ------------------------------------------------------------------
static __device__ __forceinline__
void gemm_tiles(const float* __restrict__ w, const float* inPad,
                int ocbase, int lane, v8f* acc) {
  const int hi = (lane >> 4) * 2;   // K offset for upper half-wave
  const int n  = lane & 15;
  for (int kk = 0; kk < KBLOCKS; ++kk) {
    const int k4  = kk * 4;
    const int dk  = k4 >> 8;        // kernel tap (4 | 256 -> constant per block)
    const int ic0 = k4 & 255;
    // B fragment: w[oc][ic][dk], strides (768,3,1)
    const float* wp = w + (size_t)(ocbase + n) * 768 + (size_t)(ic0 + hi) * 3 + dk;
    v2f bf; bf.x = wp[0]; bf.y = wp[3];
    #pragma unroll
    for (int m = 0; m < MTILES; ++m) {
      const float* ap = inPad + (m * 16 + (lane & 15) + dk) * HIDDEN + ic0 + hi;
      v2f af; af.x = ap[0]; af.y = ap[1];
      acc[m] = __builtin_amdgcn_wmma_f32_16x16x4_f32(
          false, af, false, bf, (short)0, acc[m], false, false);
    }
  }
}

// ------------------------------------------------------------------
// Kernel 2: per-batch conv stack (conv0 VALU, conv1..3 via f32 WMMA)
// then temporal mean -> z[b][0..255]
// ------------------------------------------------------------------
__global__ void __launch_bounds__(256)
k_encoder(const float* __restrict__ xmean,
          const float* __restrict__ w0, const float* __restrict__ b0,
          const float* __restrict__ w1, const float* __restrict__ b1,
          const float* __restrict__ w2, const float* __restrict__ b2,
          const float* __restrict__ w3, const float* __restrict__ b3,
          float* __restrict__ zout) {
  extern __shared__ float smem[];
  float* inPad = smem;                           // 210*256 f32  (padded acts)
  float* hbuf  = smem + PADROWS * HIDDEN;        // 200*128 f32  (oc 0..127 staging)
  float* zm    = hbuf + NBANDS * 128;            // 208 f32      (padded x_mean row)
  const int tid  = threadIdx.x;
  const int lane = tid & 31;
  const int wave = tid >> 5;
  const int b    = blockIdx.x;

  for (int i = tid; i < PADROWS * HIDDEN; i += 256) inPad[i] = 0.f;
  if (tid < 208) zm[tid] = 0.f;
  __syncthreads();
  if (tid >= 1 && tid <= 200) zm[tid] = xmean[(size_t)b * NBANDS + tid - 1];
  __syncthreads();

  // conv0: in_ch = 1 -> 256, one oc per thread
  {
    const int oc = tid;
    const float wA = w0[oc * 3 + 0], wB = w0[oc * 3 + 1], wC = w0[oc * 3 + 2];
    const float bb = b0[oc];
    for (int t = 0; t < NBANDS; ++t) {
      float v = bb + wA * zm[t] + wB * zm[t + 1] + wC * zm[t + 2];
      inPad[(t + 1) * HIDDEN + oc] = fmaxf(v, 0.f);
    }
  }
  __syncthreads();

  const float* Ws[3] = {w1, w2, w3};
  const float* Bs[3] = {b1, b2, b3};
  for (int layer = 0; layer < 3; ++layer) {
    const float* w    = Ws[layer];
    const float* bias = Bs[layer];

    // pass 1: oc [0,128) -> staged in hbuf (inPad still needed for pass 2)
    {
      const int ocbase = wave * 16;
      v8f acc[MTILES] = {};
      gemm_tiles(w, inPad, ocbase, lane, acc);
      const int   oc = ocbase + (lane & 15);
      const float bb = bias[oc];
      #pragma unroll
      for (int m = 0; m < MTILES; ++m) {
        #pragma unroll
        for (int r = 0; r < 8; ++r) {
          const int t = m * 16 + r + 8 * (lane >> 4);
          if (t < NBANDS) hbuf[t * 128 + oc] = fmaxf(acc[m][r] + bb, 0.f);
        }
      }
    }

    // pass 2: oc [128,256) kept in registers until inPad can be overwritten
    {
      const int ocbase = 128 + wave * 16;
      v8f acc[MTILES] = {};
      gemm_tiles(w, inPad, ocbase, lane, acc);
      __syncthreads();                      // everyone done reading inPad / writing hbuf
      const int   oc = ocbase + (lane & 15);
      const float bb = bias[oc];
      #pragma unroll
      for (int m = 0; m < MTILES; ++m) {
        #pragma unroll
        for (int r = 0; r < 8; ++r) {
          const int t = m * 16 + r + 8 * (lane >> 4);
          if (t < NBANDS) inPad[(t + 1) * HIDDEN + oc] = fmaxf(acc[m][r] + bb, 0.f);
        }
      }
      for (int i = tid; i < NBANDS * 128; i += 256)
        inPad[(1 + (i >> 7)) * HIDDEN + (i & 127)] = hbuf[i];
      __syncthreads();
    }
  }

  // temporal mean over 200 positions -> z[b][oc]
  {
    const int oc = tid;
    float s = 0.f;
    for (int p = 1; p <= NBANDS; ++p) s += inPad[p * HIDDEN + oc];
    zout[(size_t)b * HIDDEN + oc] = s * (1.0f / (float)NBANDS);
  }
}

// ------------------------------------------------------------------
// Kernel 3: VQ argmin + quantized output + decode/recon + partial losses
// ------------------------------------------------------------------
__global__ void __launch_bounds__(256)
k_vq(const float* __restrict__ zg, const float* __restrict__ codebook,
     const float* __restrict__ dec_w, const float* __restrict__ dec_b,
     const float* __restrict__ xmean, float* __restrict__ out,
     float* __restrict__ ssr, float* __restrict__ ssv) {
  __shared__ float zsh[256], qsh[256], redd[256];
  __shared__ int   redi[256];
  __shared__ int   bestIdx;
  const int tid = threadIdx.x, b = blockIdx.x;
  zsh[tid] = zg[(size_t)b * 256 + tid];
  __syncthreads();

  float best = 3.4e38f; int bidx = 0;
  for (int j = 0; j < 4; ++j) {
    const int k = tid * 4 + j;
    const float* c = codebook + (size_t)k * 256;
    float d = 0.f;
    for (int h = 0; h < 256; ++h) { float t = c[h] - zsh[h]; d += t * t; }
    if (d < best) { best = d; bidx = k; }
  }
  redd[tid] = best; redi[tid] = bidx;
  __syncthreads();
  for (int off = 128; off > 0; off >>= 1) {
    if (tid < off) {
      float d2 = redd[tid + off]; int i2 = redi[tid + off];
      if (d2 < redd[tid] || (d2 == redd[tid] && i2 < redi[tid])) {
        redd[tid] = d2; redi[tid] = i2;
      }
    }
    __syncthreads();
  }
  if (tid == 0) bestIdx = redi[0];
  __syncthreads();
  const int idx = bestIdx;

  const float q = codebook[(size_t)idx * 256 + tid];
  qsh[tid] = q;
  out[25600 + (size_t)b * 256 + tid] = q;          // quantized_st (fwd == quantized)
  const float dv = q - zsh[tid];
  redd[tid] = dv * dv;
  __syncthreads();
  for (int off = 128; off > 0; off >>= 1) {
    if (tid < off) redd[tid] += redd[tid + off];
    __syncthreads();
  }
  if (tid == 0) { ssv[b] = redd[0]; out[58368 + b] = (float)idx; }
  __syncthreads();

  float part = 0.f;
  if (tid < NBANDS) {
    const float* dw = dec_w + (size_t)tid * 256;
    float r = dec_b[tid];
    for (int h = 0; h < 256; ++h) r += dw[h] * qsh[h];
    out[(size_t)b * NBANDS + tid] = r;             // recon
    const float e = r - xmean[(size_t)b * NBANDS + tid];
    part = e * e;
  }
  redd[tid] = part;
  __syncthreads();
  for (int off = 128; off > 0; off >>= 1) {
    if (tid < off) redd[tid] += redd[tid + off];
    __syncthreads();
  }
  if (tid == 0) ssr[b] = redd[0];
}

// ------------------------------------------------------------------
// Kernel 4: final scalar loss
// ------------------------------------------------------------------
__global__ void k_loss(const float* __restrict__ ssr, const float* __restrict__ ssv,
                       float* __restrict__ out) {
  __shared__ float r1[128], r2[128];
  const int tid = threadIdx.x;
  r1[tid] = ssr[tid]; r2[tid] = ssv[tid];
  __syncthreads();
  for (int off = 64; off > 0; off >>= 1) {
    if (tid < off) { r1[tid] += r1[tid + off]; r2[tid] += r2[tid + off]; }
    __syncthreads();
  }
  if (tid == 0)
    out[58496] = r1[0] * (1.0f / 25600.0f) + 1.25f * r2[0] * (1.0f / 32768.0f);
}

// ------------------------------------------------------------------
extern "C" void kernel_launch(void* const* d_in, const int* in_sizes, int n_in,
                              void* d_out, int out_size, void* d_ws, size_t ws_size,
                              hipStream_t stream) {
  const float* x  = (const float*)d_in[0];
  const float* w0 = (const float*)d_in[1];
  const float* b0 = (const float*)d_in[2];
  const float* w1 = (const float*)d_in[3];
  const float* b1 = (const float*)d_in[4];
  const float* w2 = (const float*)d_in[5];
  const float* b2 = (const float*)d_in[6];
  const float* w3 = (const float*)d_in[7];
  const float* b3 = (const float*)d_in[8];
  const float* cb = (const float*)d_in[9];
  const float* dw = (const float*)d_in[10];
  const float* db = (const float*)d_in[11];

  float* wsf   = (float*)d_ws;
  float* xmean = wsf;                       // 25600
  float* zg    = wsf + 25600;               // 32768
  float* ssr   = wsf + 25600 + 32768;       // 128
  float* ssv   = ssr + 128;                 // 128
  float* out   = (float*)d_out;             // 25600 + 32768 + 128 + 1 = 58497

  k_xmean<<<128 * NBANDS, 256, 0, stream>>>(x, xmean);

  const size_t smem =
      (size_t)(PADROWS * HIDDEN + NBANDS * 128 + 208) * sizeof(float); // ~318 KB
  k_encoder<<<128, 256, smem, stream>>>(xmean, w0, b0, w1, b1, w2, b2, w3, b3, zg);

  k_vq<<<128, 256, 0, stream>>>(zg, cb, dw, db, xmean, out, ssr, ssv);
  k_loss<<<1, 128, 0, stream>>>(ssr, ssv, out);
}